// Discriminator_68813966016698
// MI455X (gfx1250) — compile-verified
//
#include <hip/hip_runtime.h>

typedef __attribute__((ext_vector_type(2))) float v2f;
typedef __attribute__((ext_vector_type(4))) float v4f;
typedef __attribute__((ext_vector_type(8))) float v8f;

constexpr int N    = 8192;
constexpr int F    = 128;
constexpr int E    = 262144;
constexpr int META = 64;
constexpr int NOUT = 17;   // LAM + 1

// ---------------- degree / norm ----------------
__global__ __launch_bounds__(256) void k_init_deg(float* deg) {
  int i = blockIdx.x * 256 + threadIdx.x;
  if (i < N) deg[i] = 1.0f;  // self-loop contributes 1
}

__global__ __launch_bounds__(256) void k_count_deg(const int* __restrict__ dst, float* deg) {
  int e = blockIdx.x * 256 + threadIdx.x;
  if (e < E) atomicAdd(&deg[dst[e]], 1.0f);
}

__global__ __launch_bounds__(256) void k_rsqrt_inplace(float* deg) {
  int i = blockIdx.x * 256 + threadIdx.x;
  if (i < N) deg[i] = rsqrtf(deg[i]);
}

// ---------------- layer-1 GEMM via f32 WMMA: h[N,8] = x[N,128] @ W1[128,8] ----------------
// One wave per 16-row tile; V_WMMA_F32_16X16X4_F32, K fully unrolled (32 WMMAs).
// B panel is zero-padded to 16 cols and transposed into LDS so the hot loop is
// branch-free: per step one global_load_b64 (A) + one ds_load_b64 (B) + wmma.
// A layout (32-bit A 16x4): lanes 0-15 row M=l15 hold K=k0,k0+1; lanes 16-31 hold K=k0+2,k0+3.
// B layout (4x16): lanes 0-15 rows K=k0,k0+1 at col l15; lanes 16-31 rows K=k0+2,k0+3.
// C/D: VGPR r -> row (r + 8*half), col l15.
__global__ __launch_bounds__(256) void k_gemm_w1_wmma(const float* __restrict__ x,
                                                      const float* __restrict__ W1,
                                                      float* __restrict__ h) {
  __shared__ float w1t[16 * 128];  // w1t[n][k], cols 8..15 zero
  for (int t = threadIdx.x; t < 16 * 128; t += 256) {
    int nn = t >> 7;   // 0..15
    int kk = t & 127;  // 0..127
    w1t[t] = (nn < 8) ? W1[kk * 8 + nn] : 0.0f;
  }
  __syncthreads();

  const int lane = threadIdx.x & 31;
  const int wave = threadIdx.x >> 5;
  const int tile = blockIdx.x * 8 + wave;  // 512 tiles of 16 rows
  const int half = lane >> 4;
  const int l15  = lane & 15;
  const float* xrow = x + (size_t)(tile * 16 + l15) * F + half * 2;
  const float* wrow = w1t + l15 * 128 + half * 2;

  v8f c = {};
#pragma unroll
  for (int k0 = 0; k0 < F; k0 += 4) {
    v2f a, b;
    a.x = xrow[k0];
    a.y = xrow[k0 + 1];
    b.x = wrow[k0];
    b.y = wrow[k0 + 1];
    c = __builtin_amdgcn_wmma_f32_16x16x4_f32(false, a, false, b, (short)0, c, false, false);
  }

  if (l15 < 8) {
#pragma unroll
    for (int r = 0; r < 8; ++r)
      h[(size_t)(tile * 16 + half * 8 + r) * 8 + l15] = c[r];
  }
}

// ---------------- GCN aggregation ----------------
// out[i,:] = bias + dinv[i]^2 * h[i,:]  (self-loop folded in, no atomics needed)
template <int FO>
__global__ __launch_bounds__(256) void k_agg_init(float* __restrict__ out,
                                                  const float* __restrict__ bias,
                                                  const float* __restrict__ dinv,
                                                  const float* __restrict__ h) {
  int i = blockIdx.x * 256 + threadIdx.x;
  if (i >= N) return;
  float w = dinv[i] * dinv[i];
#pragma unroll
  for (int f = 0; f < FO; ++f)
    out[i * FO + f] = bias[f] + w * h[i * FO + f];
}

// out[dst,:] += dinv[src]*dinv[dst] * h[src,:]
template <int FO>
__global__ __launch_bounds__(256) void k_agg_edges(const int* __restrict__ src,
                                                   const int* __restrict__ dst,
                                                   const float* __restrict__ dinv,
                                                   const float* __restrict__ h,
                                                   float* __restrict__ out) {
  int e = blockIdx.x * 256 + threadIdx.x;
  if (e >= E) return;
  int s = src[e], d = dst[e];
  float nrm = dinv[s] * dinv[d];
#pragma unroll
  for (int f = 0; f < FO; ++f)
    atomicAdd(&out[d * FO + f], nrm * h[s * FO + f]);
}

// tiny dense layers, per node (no bias; bias applied post-aggregation)
template <int FI, int FO>
__global__ __launch_bounds__(256) void k_dense(const float* __restrict__ x,
                                               const float* __restrict__ W,
                                               float* __restrict__ out) {
  int i = blockIdx.x * 256 + threadIdx.x;
  if (i >= N) return;
  float xi[FI];
#pragma unroll
  for (int f = 0; f < FI; ++f) xi[f] = x[i * FI + f];
#pragma unroll
  for (int o = 0; o < FO; ++o) {
    float acc = 0.0f;
#pragma unroll
    for (int f = 0; f < FI; ++f) acc = fmaf(xi[f], W[f * FO + o], acc);
    out[i * FO + o] = acc;
  }
}

// ---------------- Wi GEMV (bandwidth-dominant: 512 MB stream) ----------------
__global__ __launch_bounds__(256) void k_init_bi(const float* __restrict__ bi, float* r) {
  int j = blockIdx.x * 256 + threadIdx.x;
  if (j < N) r[j] = bi[j];
}

// grid (8 column-blocks of 1024 cols, 64 i-chunks of 256 rows).
// Each thread owns 4 consecutive columns -> global_load_b128 stream.
__global__ __launch_bounds__(256) void k_gemv_wi(const float* __restrict__ h1,
                                                 const float* __restrict__ h2,
                                                 const float* __restrict__ Wi,
                                                 float* __restrict__ r) {
  __shared__ float cs[256];
  int j4 = (blockIdx.x * 256 + threadIdx.x) * 4;  // first of 4 columns
  int i0 = blockIdx.y * 256;                      // input-row chunk
  int it = i0 + threadIdx.x;
  cs[threadIdx.x] = (it < N) ? h1[it] : h2[it - N];
  __syncthreads();
  v4f acc = {};
#pragma unroll 4
  for (int ii = 0; ii < 256; ++ii) {
    const v4f w = *(const v4f*)(Wi + (size_t)(i0 + ii) * N + j4);
    float ci = cs[ii];
    acc.x = fmaf(ci, w.x, acc.x);
    acc.y = fmaf(ci, w.y, acc.y);
    acc.z = fmaf(ci, w.z, acc.z);
    acc.w = fmaf(ci, w.w, acc.w);
  }
  atomicAdd(&r[j4 + 0], acc.x);
  atomicAdd(&r[j4 + 1], acc.y);
  atomicAdd(&r[j4 + 2], acc.z);
  atomicAdd(&r[j4 + 3], acc.w);
}

// ---------------- final Wf head: out[k] = bf[k] + concat(r, meta) . Wf[:,k] ----------------
__global__ __launch_bounds__(256) void k_final_wf(const float* __restrict__ r,
                                                  const float* __restrict__ meta,
                                                  const float* __restrict__ Wf,
                                                  const float* __restrict__ bf,
                                                  float* __restrict__ out) {
  int k = blockIdx.x;  // 0..16
  float acc = 0.0f;
  for (int i = threadIdx.x; i < N + META; i += 256) {
    float ci = (i < N) ? r[i] : meta[i - N];
    acc = fmaf(ci, Wf[i * NOUT + k], acc);
  }
  __shared__ float red[256];
  red[threadIdx.x] = acc;
  __syncthreads();
  for (int s = 128; s > 0; s >>= 1) {
    if (threadIdx.x < s) red[threadIdx.x] += red[threadIdx.x + s];
    __syncthreads();
  }
  if (threadIdx.x == 0) out[k] = red[0] + bf[k];
}

// ---------------- branch driver (host-side helper; only launches on stream) ----------------
static void run_branch(const float* x, const int* ei,
                       const float* W1, const float* b1,
                       const float* W2, const float* b2,
                       const float* W3, const float* b3,
                       const float* W4, const float* b4,
                       float* dinv, float* bufA, float* bufB, float* hv,
                       hipStream_t stream) {
  const int* src = ei;
  const int* dst = ei + E;
  k_init_deg<<<N / 256, 256, 0, stream>>>(dinv);
  k_count_deg<<<E / 256, 256, 0, stream>>>(dst, dinv);
  k_rsqrt_inplace<<<N / 256, 256, 0, stream>>>(dinv);

  // layer 1: WMMA GEMM then aggregate
  k_gemm_w1_wmma<<<N / 128, 256, 0, stream>>>(x, W1, bufA);
  k_agg_init<8><<<N / 256, 256, 0, stream>>>(bufB, b1, dinv, bufA);
  k_agg_edges<8><<<E / 256, 256, 0, stream>>>(src, dst, dinv, bufA, bufB);

  // layer 2: 8 -> 4
  k_dense<8, 4><<<N / 256, 256, 0, stream>>>(bufB, W2, bufA);
  k_agg_init<4><<<N / 256, 256, 0, stream>>>(bufB, b2, dinv, bufA);
  k_agg_edges<4><<<E / 256, 256, 0, stream>>>(src, dst, dinv, bufA, bufB);

  // layer 3: 4 -> 2
  k_dense<4, 2><<<N / 256, 256, 0, stream>>>(bufB, W3, bufA);
  k_agg_init<2><<<N / 256, 256, 0, stream>>>(bufB, b3, dinv, bufA);
  k_agg_edges<2><<<E / 256, 256, 0, stream>>>(src, dst, dinv, bufA, bufB);

  // layer 4: 2 -> 1 (result goes to hv[N])
  k_dense<2, 1><<<N / 256, 256, 0, stream>>>(bufB, W4, bufA);
  k_agg_init<1><<<N / 256, 256, 0, stream>>>(hv, b4, dinv, bufA);
  k_agg_edges<1><<<E / 256, 256, 0, stream>>>(src, dst, dinv, bufA, hv);
}

extern "C" void kernel_launch(void* const* d_in, const int* in_sizes, int n_in,
                              void* d_out, int out_size, void* d_ws, size_t ws_size,
                              hipStream_t stream) {
  const float* x1   = (const float*)d_in[0];
  const float* x2   = (const float*)d_in[1];
  const float* meta = (const float*)d_in[2];
  const float* W1   = (const float*)d_in[3];
  const float* b1   = (const float*)d_in[4];
  const float* W2   = (const float*)d_in[5];
  const float* b2   = (const float*)d_in[6];
  const float* W3   = (const float*)d_in[7];
  const float* b3   = (const float*)d_in[8];
  const float* W4   = (const float*)d_in[9];
  const float* b4   = (const float*)d_in[10];
  const float* Wi   = (const float*)d_in[11];
  const float* bi   = (const float*)d_in[12];
  const float* Wf   = (const float*)d_in[13];
  const float* bf   = (const float*)d_in[14];
  const int*   ei1  = (const int*)d_in[15];
  const int*   ei2  = (const int*)d_in[16];
  float* out = (float*)d_out;

  // workspace layout (bytes)
  char* ws = (char*)d_ws;
  float* dinv   = (float*)(ws);                     //  32 KB
  float* bufA   = (float*)(ws + 32768);             // 256 KB  (h buffers [N,8])
  float* bufB   = (float*)(ws + 32768 + 262144);    // 256 KB
  float* h1v    = (float*)(ws + 557056);            //  32 KB
  float* h2v    = (float*)(ws + 589824);            //  32 KB
  float* rinter = (float*)(ws + 622592);            //  32 KB

  run_branch(x1, ei1, W1, b1, W2, b2, W3, b3, W4, b4, dinv, bufA, bufB, h1v, stream);
  run_branch(x2, ei2, W1, b1, W2, b2, W3, b3, W4, b4, dinv, bufA, bufB, h2v, stream);

  // result_inter = concat(h1,h2) @ Wi + bi   (512 MB weight stream — BW bound)
  k_init_bi<<<N / 256, 256, 0, stream>>>(bi, rinter);
  dim3 g(N / 1024, 64);
  k_gemv_wi<<<g, 256, 0, stream>>>(h1v, h2v, Wi, rinter);

  // out = concat(result_inter, meta) @ Wf + bf
  k_final_wf<<<NOUT, 256, 0, stream>>>(rinter, meta, Wf, bf, out);
}